// LowHighQuantizer_71665824301782
// MI455X (gfx1250) — compile-verified
//
#include <hip/hip_runtime.h>
#include <stdint.h>

// ---------------- problem constants ----------------
#define ROWS 4096
#define COLS 11008
#define NTOT 45088768LL            // ROWS*COLS
#define N4   11272192LL            // NTOT/4
#define RANK_LO 2254437u           // 0-based rank of low threshold
#define RANK_HI 42834329u          // 0-based rank of high threshold

#define H1_BINS 16384              // top-14 bits of sortable key
#define H2_BINS 262144             // low-18 bits refinement

// workspace layout (bytes)
#define WS_H1   0
#define WS_H2LO 65536
#define WS_H2HI (65536 + 1048576)
#define WS_CTRL (65536 + 2097152)
#define WS_ZERO_BYTES (WS_CTRL + 64)

typedef unsigned int u32;
typedef unsigned long long u64;
typedef u32 u32x4 __attribute__((ext_vector_type(4)));
typedef u32 u32x8 __attribute__((ext_vector_type(8)));

// monotone float -> u32 key (ascending float order == ascending unsigned order)
__device__ __forceinline__ u32 f2key(float f) {
  u32 b = __float_as_uint(f);
  return b ^ ((b & 0x80000000u) ? 0xFFFFFFFFu : 0x80000000u);
}
__device__ __forceinline__ float key2f(u32 k) {
  u32 b = (k & 0x80000000u) ? (k ^ 0x80000000u) : ~k;
  return __uint_as_float(b);
}

// ---------------- pass 1: top-14-bit histogram (LDS ds_add atomics) ----------------
__global__ __launch_bounds__(256) void k_hist1(const float4* __restrict__ x,
                                               u32* __restrict__ h1) {
  __shared__ u32 lh[H1_BINS];
  for (int i = threadIdx.x; i < H1_BINS; i += blockDim.x) lh[i] = 0u;
  __syncthreads();
  long long stride = (long long)gridDim.x * blockDim.x;
  for (long long i = (long long)blockIdx.x * blockDim.x + threadIdx.x; i < N4; i += stride) {
    float4 v = x[i];
    atomicAdd(&lh[f2key(v.x) >> 18], 1u);
    atomicAdd(&lh[f2key(v.y) >> 18], 1u);
    atomicAdd(&lh[f2key(v.z) >> 18], 1u);
    atomicAdd(&lh[f2key(v.w) >> 18], 1u);
  }
  __syncthreads();
  for (int i = threadIdx.x; i < H1_BINS; i += blockDim.x) {
    u32 c = lh[i];
    if (c) atomicAdd(&h1[i], c);
  }
}

// ---------------- pass 2: scan to find candidate buckets ----------------
__global__ __launch_bounds__(256) void k_scan1(const u32* __restrict__ h1,
                                               u32* __restrict__ ctrl) {
  __shared__ u32 part[256];
  __shared__ u32 pfx[256];
  const int t = threadIdx.x;
  const int per = H1_BINS / 256;  // 64
  u32 s = 0;
  for (int i = 0; i < per; i++) s += h1[t * per + i];
  part[t] = s;
  __syncthreads();
  if (t == 0) {
    u32 a = 0;
    for (int i = 0; i < 256; i++) { pfx[i] = a; a += part[i]; }
  }
  __syncthreads();
  u32 cum = pfx[t];
  for (int i = 0; i < per; i++) {
    u32 c = h1[t * per + i];
    if (RANK_LO >= cum && RANK_LO < cum + c) { ctrl[0] = (u32)(t * per + i); ctrl[1] = RANK_LO - cum; }
    if (RANK_HI >= cum && RANK_HI < cum + c) { ctrl[2] = (u32)(t * per + i); ctrl[3] = RANK_HI - cum; }
    cum += c;
  }
}

// ---------------- pass 3: low-18-bit refinement histograms ----------------
__global__ __launch_bounds__(256) void k_hist2(const float4* __restrict__ x,
                                               const u32* __restrict__ ctrl,
                                               u32* __restrict__ h2lo,
                                               u32* __restrict__ h2hi) {
  const u32 blo = ctrl[0], bhi = ctrl[2];
  long long stride = (long long)gridDim.x * blockDim.x;
  for (long long i = (long long)blockIdx.x * blockDim.x + threadIdx.x; i < N4; i += stride) {
    float4 v = x[i];
    u32 k0 = f2key(v.x), k1 = f2key(v.y), k2 = f2key(v.z), k3 = f2key(v.w);
    if ((k0 >> 18) == blo) atomicAdd(&h2lo[k0 & 0x3FFFFu], 1u);
    if ((k1 >> 18) == blo) atomicAdd(&h2lo[k1 & 0x3FFFFu], 1u);
    if ((k2 >> 18) == blo) atomicAdd(&h2lo[k2 & 0x3FFFFu], 1u);
    if ((k3 >> 18) == blo) atomicAdd(&h2lo[k3 & 0x3FFFFu], 1u);
    if ((k0 >> 18) == bhi) atomicAdd(&h2hi[k0 & 0x3FFFFu], 1u);
    if ((k1 >> 18) == bhi) atomicAdd(&h2hi[k1 & 0x3FFFFu], 1u);
    if ((k2 >> 18) == bhi) atomicAdd(&h2hi[k2 & 0x3FFFFu], 1u);
    if ((k3 >> 18) == bhi) atomicAdd(&h2hi[k3 & 0x3FFFFu], 1u);
  }
}

// ---------------- pass 4: exact threshold recovery ----------------
__global__ __launch_bounds__(256) void k_scan2(const u32* __restrict__ h2lo,
                                               const u32* __restrict__ h2hi,
                                               u32* __restrict__ ctrl) {
  const int which = blockIdx.x;  // 0 = low threshold, 1 = high threshold
  const u32* h = which ? h2hi : h2lo;
  const u32 bucket = ctrl[which ? 2 : 0];
  const u32 rank   = ctrl[which ? 3 : 1];
  __shared__ u32 part[256];
  __shared__ u32 pfx[256];
  const int t = threadIdx.x;
  const int per = H2_BINS / 256;  // 1024
  u32 s = 0;
  for (int i = 0; i < per; i++) s += h[t * per + i];
  part[t] = s;
  __syncthreads();
  if (t == 0) {
    u32 a = 0;
    for (int i = 0; i < 256; i++) { pfx[i] = a; a += part[i]; }
  }
  __syncthreads();
  u32 cum = pfx[t];
  for (int i = 0; i < per; i++) {
    u32 c = h[t * per + i];
    if (rank >= cum && rank < cum + c) {
      u32 key = (bucket << 18) | (u32)(t * per + i);
      ctrl[4 + which] = __float_as_uint(key2f(key));
    }
    cum += c;
  }
}

// ---------------- pass 5: apply (TDM-staged streaming pass) ----------------
#define CHUNK 5504                 // half a row, floats (22016 B tile)
#define CPB   16                   // chunks per block (8 rows); grid = 8192/16 = 512

// Issue a 1-D Tensor Data Mover load of `nelem` fp32 from gptr into LDS offset.
// D# groups per CDNA5 ISA ch. 8 (CDNA5 ISA). Tile described as its own
// tensor (fully in-bounds), data_size=4B, no padding, no iterate, no multicast.
__device__ __forceinline__ void tdm_load(const float* gptr, u32 lds_off, u32 nelem) {
  u64 ga = (u64)(size_t)gptr;
  u32x4 g0;
  g0[0] = 1u;                                             // count=1, user descriptor
  g0[1] = lds_off;                                        // lds_addr (bytes)
  g0[2] = (u32)ga;                                        // global_addr[31:0]
  g0[3] = (u32)((ga >> 32) & 0x01FFFFFFu) | (2u << 30);   // global_addr[56:32] | type=2
  u32x8 g1;
  g1[0] = 0x00020000u;                 // workgroup_mask=0, data_size=2 (4 bytes)
  g1[1] = (nelem & 0xFFFFu) << 16;     // tensor_dim0[15:0] at bits[63:48]
  g1[2] = (nelem >> 16) | (1u << 16);  // tensor_dim0[31:16] | tensor_dim1=1
  g1[3] = (nelem & 0xFFFFu) << 16;     // tile_dim0 at bits[127:112]
  g1[4] = 1u;                          // tile_dim1=1, tile_dim2=0
  g1[5] = nelem;                       // tensor_dim0_stride[31:0]
  g1[6] = 0u;                          // stride0[47:32] | stride1[15:0]
  g1[7] = 0u;
  asm volatile("tensor_load_to_lds %0, %1" :: "s"(g0), "s"(g1) : "memory");
}

__device__ __forceinline__ float qdq(float v, float s, float z, float maxq) {
  float q = rintf(v / s) + z;                 // round-to-nearest-even == jnp.round
  q = fminf(fmaxf(q, 0.0f), maxq);
  return s * (q - z);
}
__device__ __forceinline__ float elem(float v, float tlo, float thi,
                                      float s1, float z1, float s2, float z2) {
  bool bulk = (v > tlo) && (v < thi);
  float a = bulk ? v : 0.0f;
  float b = bulk ? 0.0f : v;
  return qdq(a, s1, z1, 1.0f) + qdq(b, s2, z2, 255.0f);
}

__global__ __launch_bounds__(256) void k_apply(const float* __restrict__ x,
                                               const float* __restrict__ sl,
                                               const float* __restrict__ zl,
                                               const float* __restrict__ sh,
                                               const float* __restrict__ zh,
                                               const u32* __restrict__ ctrl,
                                               float* __restrict__ out) {
  __shared__ float buf[2][CHUNK];
  const float tlo = __uint_as_float(ctrl[4]);
  const float thi = __uint_as_float(ctrl[5]);
  const long long c0 = (long long)blockIdx.x * CPB;
  const bool issuer = (threadIdx.x < 32);   // wave 0 drives the TDM

  if (issuer) tdm_load(x + c0 * CHUNK, (u32)(size_t)&buf[0][0], CHUNK);

  for (int c = 0; c < CPB; c++) {
    long long chunk = c0 + c;
    int row = (int)(chunk >> 1);            // 2 chunks per row
    if (issuer) {
      if (c + 1 < CPB) {
        tdm_load(x + (chunk + 1) * CHUNK, (u32)(size_t)&buf[(c + 1) & 1][0], CHUNK);
        __builtin_amdgcn_s_wait_tensorcnt(1);   // oldest (current chunk) done
      } else {
        __builtin_amdgcn_s_wait_tensorcnt(0);
      }
    }
    __syncthreads();                         // LDS tile visible to all waves

    float s1 = sl[row], z1 = zl[row], s2 = sh[row], z2 = zh[row];
    const float4* src = (const float4*)&buf[c & 1][0];
    float4* dst = (float4*)(out + chunk * CHUNK);
    for (int i = threadIdx.x; i < CHUNK / 4; i += 256) {
      float4 v = src[i];
      float4 o;
      o.x = elem(v.x, tlo, thi, s1, z1, s2, z2);
      o.y = elem(v.y, tlo, thi, s1, z1, s2, z2);
      o.z = elem(v.z, tlo, thi, s1, z1, s2, z2);
      o.w = elem(v.w, tlo, thi, s1, z1, s2, z2);
      dst[i] = o;
    }
    __syncthreads();                         // done reading before buffer reuse
  }
}

// ---------------- host-side launcher ----------------
extern "C" void kernel_launch(void* const* d_in, const int* in_sizes, int n_in,
                              void* d_out, int out_size, void* d_ws, size_t ws_size,
                              hipStream_t stream) {
  const float* x  = (const float*)d_in[0];
  const float* sl = (const float*)d_in[1];
  const float* zl = (const float*)d_in[2];
  const float* sh = (const float*)d_in[3];
  const float* zh = (const float*)d_in[4];
  float* out = (float*)d_out;

  unsigned char* ws = (unsigned char*)d_ws;
  u32* h1   = (u32*)(ws + WS_H1);
  u32* h2lo = (u32*)(ws + WS_H2LO);
  u32* h2hi = (u32*)(ws + WS_H2HI);
  u32* ctrl = (u32*)(ws + WS_CTRL);

  hipMemsetAsync(ws, 0, WS_ZERO_BYTES, stream);   // graph-capturable memset node

  k_hist1<<<512, 256, 0, stream>>>((const float4*)x, h1);
  k_scan1<<<1, 256, 0, stream>>>(h1, ctrl);
  k_hist2<<<2048, 256, 0, stream>>>((const float4*)x, ctrl, h2lo, h2hi);
  k_scan2<<<2, 256, 0, stream>>>(h2lo, h2hi, ctrl);
  k_apply<<<512, 256, 0, stream>>>(x, sl, zl, sh, zh, ctrl, out);
}